// Sonic_22041772163364
// MI455X (gfx1250) — compile-verified
//
#include <hip/hip_runtime.h>
#include <math.h>
#include <stdint.h>

// ---------------- problem constants ----------------
constexpr int BATCH  = 8;
constexpr int C_IN   = 16;
constexpr int M_MODES= 32;
constexpr int K_OUT  = 64;
constexpr int H      = 384;
constexpr int W      = 384;
constexpr int WF     = W/2 + 1;          // 193 rfft bins
constexpr int NF     = H * WF;           // 74112 frequency points
constexpr int WTILES = (WF + 7) / 8;     // 25 column tiles of 8
constexpr int LDSW   = 9;                // padded LDS row stride (bank spread)
constexpr float TWO_PI = 6.28318530717958647692f;

typedef float v2f __attribute__((ext_vector_type(2)));
typedef float v8f __attribute__((ext_vector_type(8)));

__device__ inline v8f wmma4(v2f a, v2f b, v8f c) {
    // V_WMMA_F32_16X16X4_F32 : D = A(16x4) * B(4x16) + C(16x16)
    return __builtin_amdgcn_wmma_f32_16x16x4_f32(
        /*neg_a=*/false, a, /*neg_b=*/false, b,
        /*c_mod=*/(short)0, c, /*reuse_a=*/false, /*reuse_b=*/false);
}

__device__ inline float2 cmulf(float2 a, float2 b) {
    return make_float2(a.x*b.x - a.y*b.y, a.x*b.y + a.y*b.x);
}

// ---------------- CDNA5 async global<->LDS movers (ASYNCcnt) ----------------
__device__ inline void async_load_b64(uint32_t lds_addr, const void* gaddr) {
    asm volatile("global_load_async_to_lds_b64 %0, %1, off"
                 :: "v"(lds_addr), "v"((uint64_t)(uintptr_t)gaddr)
                 : "memory");
}
__device__ inline void async_store_b64(const void* gaddr, uint32_t lds_addr) {
    asm volatile("global_store_async_from_lds_b64 %0, %1, off"
                 :: "v"((uint64_t)(uintptr_t)gaddr), "v"(lds_addr)
                 : "memory");
}
__device__ inline void wait_async0() {
    asm volatile("s_wait_asynccnt 0x0" ::: "memory");
}

// ---------------- mixed-radix 384 = 2^7 * 3 Stockham (DIF, autosort) -------
// Elements live at buf[i * st] (st = LDS row stride in float2 units).
// sgn = -1 forward, +1 inverse (unnormalized). Result ends up in bufA.
__device__ void fft384_s(float2* bufA, float2* bufB, int st,
                         int w, int nw, float sgn) {
    float2* src = bufA;
    float2* dst = bufB;
    int n = 384, s = 1;
    // 7 radix-2 stages: 192 butterflies each
    #pragma unroll 1
    for (int stg = 0; stg < 7; ++stg) {
        float theta0 = sgn * TWO_PI / (float)n;
        for (int j = w; j < 192; j += nw) {
            int p = j / s;
            int q = j - p * s;
            float2 x0 = src[(q + s * p) * st];
            float2 x1 = src[(q + s * (p + (n >> 1))) * st];
            float sn, cs;
            __sincosf(theta0 * (float)p, &sn, &cs);
            float2 wv = make_float2(cs, sn);
            float2 su = make_float2(x0.x + x1.x, x0.y + x1.y);
            float2 df = make_float2(x0.x - x1.x, x0.y - x1.y);
            dst[(q + s * (2 * p)) * st]     = su;
            dst[(q + s * (2 * p + 1)) * st] = cmulf(df, wv);
        }
        __syncthreads();
        float2* t = src; src = dst; dst = t;
        n >>= 1; s <<= 1;
    }
    // final radix-3 stage: n = 3, s = 128 (reads bufB, writes bufA)
    {
        const float c3 = -0.5f;
        const float s3 = sgn * 0.86602540378443864676f;
        for (int q = w; q < 128; q += nw) {
            float2 x0 = src[q * st];
            float2 x1 = src[(q + 128) * st];
            float2 x2 = src[(q + 256) * st];
            float2 t1 = make_float2(x1.x + x2.x, x1.y + x2.y);
            float2 t2 = make_float2(x1.x - x2.x, x1.y - x2.y);
            float zr = x0.x + c3 * t1.x;
            float zi = x0.y + c3 * t1.y;
            dst[q * st]         = make_float2(x0.x + t1.x, x0.y + t1.y);
            dst[(q + 128) * st] = make_float2(zr - s3 * t2.y, zi + s3 * t2.x);
            dst[(q + 256) * st] = make_float2(zr + s3 * t2.y, zi - s3 * t2.x);
        }
        __syncthreads();
    }
}

// ---------------- K1: per-sample mean / rstd ----------------
__global__ __launch_bounds__(256)
void stats_kernel(const float* __restrict__ x, float* __restrict__ stats) {
    const int b = blockIdx.x;
    const size_t n = (size_t)C_IN * H * W;
    const float* xb = x + (size_t)b * n;
    float s = 0.f, s2 = 0.f;
    for (size_t i = threadIdx.x; i < n; i += blockDim.x) {
        float v = xb[i]; s += v; s2 += v * v;
    }
    __shared__ float rs[256], rq[256];
    rs[threadIdx.x] = s; rq[threadIdx.x] = s2;
    __syncthreads();
    for (int o = 128; o > 0; o >>= 1) {
        if (threadIdx.x < o) { rs[threadIdx.x] += rs[threadIdx.x + o];
                               rq[threadIdx.x] += rq[threadIdx.x + o]; }
        __syncthreads();
    }
    if (threadIdx.x == 0) {
        float mean = rs[0] / (float)n;
        float var  = rq[0] / (float)n - mean * mean;
        stats[b]         = mean;
        stats[BATCH + b] = rsqrtf(var + 1e-5f);
    }
}

// ---------------- K2: normalize + forward row rFFT ----------------
__global__ __launch_bounds__(128)
void rowfft_fwd_kernel(const float* __restrict__ x, const float* __restrict__ stats,
                       float2* __restrict__ Xf) {
    __shared__ float2 bufA[384], bufB[384];
    const int row = blockIdx.x;                 // (b*C + c)*H + h
    const int b = row / (C_IN * H);
    const float mean = stats[b], rstd = stats[BATCH + b];
    const float* xr = x + (size_t)row * W;
    for (int i = threadIdx.x; i < W; i += blockDim.x)
        bufA[i] = make_float2((xr[i] - mean) * rstd, 0.f);
    __syncthreads();
    fft384_s(bufA, bufB, 1, threadIdx.x, blockDim.x, -1.f);
    float2* out = Xf + (size_t)row * WF;
    for (int i = threadIdx.x; i < WF; i += blockDim.x)
        out[i] = bufA[i];
}

// ---------------- K3/K5: column FFT, 8 columns per block ------------------
// Coalesced: a [384 x 8] float2 tile is streamed into LDS with async DMA
// (64B-contiguous per row), 8 independent column FFTs run in padded LDS,
// then the tile is scattered back with async stores.
__global__ __launch_bounds__(256)
void colfft_kernel(float2* __restrict__ data, float sgn, float scale) {
    __shared__ float2 bufA[384][LDSW], bufB[384][LDSW];
    const int img   = blockIdx.x / WTILES;
    const int wtile = blockIdx.x - img * WTILES;
    const int wf0   = wtile * 8;
    float2* base = data + (size_t)img * NF + wf0;

    // async coalesced gather of strided tile into LDS (bypasses VGPRs)
    for (int l = threadIdx.x; l < 384 * 8; l += 256) {
        int row = l >> 3, c = l & 7;
        if (wf0 + c < WF)
            async_load_b64((uint32_t)(uintptr_t)&bufA[row][c],
                           base + (size_t)row * WF + c);
    }
    wait_async0();
    __syncthreads();

    const int w = threadIdx.x >> 3;    // worker 0..31 within a column
    const int c = threadIdx.x & 7;     // column 0..7
    fft384_s(&bufA[0][c], &bufB[0][c], LDSW, w, 32, sgn);

    // apply scale in LDS (async DMA cannot scale), then async scatter back
    for (int l = threadIdx.x; l < 384 * 8; l += 256) {
        int row = l >> 3, cc = l & 7;
        float2 v = bufA[row][cc];
        bufA[row][cc] = make_float2(v.x * scale, v.y * scale);
    }
    __syncthreads();
    for (int l = threadIdx.x; l < 384 * 8; l += 256) {
        int row = l >> 3, cc = l & 7;
        if (wf0 + cc < WF)
            async_store_b64(base + (size_t)row * WF + cc,
                            (uint32_t)(uintptr_t)&bufA[row][cc]);
    }
    wait_async0();
}

// ---------------- K4: WMMA spectral mix  Yf = C * (T .* (B * Xf)) --------
// One wave handles a tile of 16 contiguous frequency points (all 8 batches).
__global__ __launch_bounds__(256)
void spectral_kernel(const float2* __restrict__ Xf, float2* __restrict__ Yf,
                     const float* __restrict__ theta,
                     const float* __restrict__ B_re, const float* __restrict__ B_im,
                     const float* __restrict__ C_re, const float* __restrict__ C_im,
                     const float* __restrict__ alpha_raw, const float* __restrict__ beta,
                     const float* __restrict__ log_dc,  const float* __restrict__ log_tau,
                     const float* __restrict__ log_scale, const float* __restrict__ log_bw) {
    __shared__ float2 zsh[8][M_MODES][16];      // per-wave Z staging (32 KB)

    const int lane  = threadIdx.x & 31;
    const int wave  = threadIdx.x >> 5;
    const int ftile = blockIdx.x * 8 + wave;
    const int f0    = ftile * 16;
    const int n     = lane & 15;                // column (frequency within tile)
    const int half  = (lane >> 4) & 1;          // lane-half selects K / M offset

    // ---- hoisted A-operands for stage A: Bc[32x16], 2 M-tiles x 4 k-steps
    v2f aBr[2][4], aBi[2][4], aBin[2][4];
    #pragma unroll
    for (int t = 0; t < 2; ++t)
        #pragma unroll
        for (int kk = 0; kk < 4; ++kk) {
            int m  = t * 16 + (lane & 15);
            int k0 = 4 * kk + 2 * half;
            aBr[t][kk].x = B_re[m * C_IN + k0];
            aBr[t][kk].y = B_re[m * C_IN + k0 + 1];
            aBi[t][kk].x = B_im[m * C_IN + k0];
            aBi[t][kk].y = B_im[m * C_IN + k0 + 1];
            aBin[t][kk].x = -aBi[t][kk].x;
            aBin[t][kk].y = -aBi[t][kk].y;
        }

    // ---- per-lane transfer function T (matches C/D accumulator layout)
    float Tre[2][8], Tim[2][8];
    {
        const int f  = f0 + n;
        const int h  = f / WF;
        const int wf = f - h * WF;
        const float kx = TWO_PI * (float)wf / (float)W;
        const float ky = TWO_PI * (float)((h < H / 2) ? h : h - H) / (float)H;
        const float w2 = kx * kx + ky * ky;
        #pragma unroll
        for (int t = 0; t < 2; ++t)
            #pragma unroll
            for (int i = 0; i < 8; ++i) {
                int m = t * 16 + i + 8 * half;
                float st, ct; __sincosf(theta[m], &st, &ct);
                float vdot = ct * kx + st * ky;
                float wperp2 = w2 - vdot * vdot;
                float sp   = log1pf(expf(alpha_raw[m]));   // softplus
                float sv   = expf(log_scale[m]);
                float tauv = expf(log_tau[m]);
                float dcv  = expf(log_dc[m]);
                float wcv  = expf(log_bw[m]);
                float Dre = sp + tauv * wperp2;
                float Dim = sv * vdot - beta[m];
                float r2 = w2 / (wcv * wcv);
                float bt = 1.0f + r2 * r2 * r2 * r2;       // Butterworth order 4
                Dre *= bt; Dim *= bt;
                float inv = dcv / (Dre * Dre + Dim * Dim);
                Tre[t][i] =  Dre * inv;
                Tim[t][i] = -Dim * inv;
            }
    }

    const v8f vzero = {0.f,0.f,0.f,0.f,0.f,0.f,0.f,0.f};

    #pragma unroll 1
    for (int b = 0; b < BATCH; ++b) {
        // ---------- stage A: Z = Bc * Xf[b]  (complex 32x16 = [32x16]*[16x16])
        v8f Zr[2], Zi[2];
        Zr[0] = vzero; Zr[1] = vzero; Zi[0] = vzero; Zi[1] = vzero;
        #pragma unroll
        for (int kk = 0; kk < 4; ++kk) {
            int c0 = 4 * kk + 2 * half;
            size_t base = ((size_t)(b * C_IN + c0)) * NF + f0 + n;
            float2 xv0 = Xf[base];
            float2 xv1 = Xf[base + NF];
            v2f bXr, bXi;
            bXr.x = xv0.x; bXr.y = xv1.x;
            bXi.x = xv0.y; bXi.y = xv1.y;
            #pragma unroll
            for (int t = 0; t < 2; ++t) {
                Zr[t] = wmma4(aBr[t][kk],  bXr, Zr[t]);
                Zr[t] = wmma4(aBin[t][kk], bXi, Zr[t]);   // - Bi*Xi
                Zi[t] = wmma4(aBr[t][kk],  bXi, Zi[t]);
                Zi[t] = wmma4(aBi[t][kk],  bXr, Zi[t]);
            }
        }
        // ---------- apply T elementwise in accumulator layout, stage to LDS
        #pragma unroll
        for (int t = 0; t < 2; ++t)
            #pragma unroll
            for (int i = 0; i < 8; ++i) {
                int m = t * 16 + i + 8 * half;
                float zr = Zr[t][i], zi = Zi[t][i];
                zsh[wave][m][n] = make_float2(zr * Tre[t][i] - zi * Tim[t][i],
                                              zr * Tim[t][i] + zi * Tre[t][i]);
            }
        __syncthreads();

        // ---------- stage B: Yf[b] = Cc * Z   ([64x32]*[32x16] complex)
        v8f Yr[4], Yi[4];
        #pragma unroll
        for (int t = 0; t < 4; ++t) { Yr[t] = vzero; Yi[t] = vzero; }
        #pragma unroll
        for (int kk = 0; kk < 8; ++kk) {
            int m0 = 4 * kk + 2 * half;
            float2 z0 = zsh[wave][m0][n];
            float2 z1 = zsh[wave][m0 + 1][n];
            v2f bZr, bZi;
            bZr.x = z0.x; bZr.y = z1.x;
            bZi.x = z0.y; bZi.y = z1.y;
            #pragma unroll
            for (int t = 0; t < 4; ++t) {
                int row = t * 16 + (lane & 15);
                v2f aCr, aCi, aCin;
                aCr.x = C_re[row * M_MODES + m0];
                aCr.y = C_re[row * M_MODES + m0 + 1];
                aCi.x = C_im[row * M_MODES + m0];
                aCi.y = C_im[row * M_MODES + m0 + 1];
                aCin.x = -aCi.x; aCin.y = -aCi.y;
                Yr[t] = wmma4(aCr,  bZr, Yr[t]);
                Yr[t] = wmma4(aCin, bZi, Yr[t]);
                Yi[t] = wmma4(aCr,  bZi, Yi[t]);
                Yi[t] = wmma4(aCi,  bZr, Yi[t]);
            }
        }
        // ---------- write Yf[b,k,f0+n]
        #pragma unroll
        for (int t = 0; t < 4; ++t)
            #pragma unroll
            for (int i = 0; i < 8; ++i) {
                int k = t * 16 + i + 8 * half;
                Yf[((size_t)(b * K_OUT + k)) * NF + f0 + n] =
                    make_float2(Yr[t][i], Yi[t][i]);
            }
        __syncthreads();   // protect zsh before next batch overwrites it
    }
}

// ---------------- K6: inverse row rFFT (Hermitian reconstruct) -----------
__global__ __launch_bounds__(128)
void rowfft_inv_kernel(const float2* __restrict__ Yf, float* __restrict__ y) {
    __shared__ float2 bufA[384], bufB[384];
    const int row = blockIdx.x;                 // (b*K + k)*H + h
    const float2* in = Yf + (size_t)row * WF;
    for (int i = threadIdx.x; i < W; i += blockDim.x) {
        float2 v;
        if (i < WF) v = in[i];
        else { float2 s = in[W - i]; v = make_float2(s.x, -s.y); }
        bufA[i] = v;
    }
    __syncthreads();
    fft384_s(bufA, bufB, 1, threadIdx.x, blockDim.x, +1.f);
    const float inv = 1.0f / 384.0f;
    float* out = y + (size_t)row * W;
    for (int i = threadIdx.x; i < W; i += blockDim.x)
        out[i] = bufA[i].x * inv;
}

// ---------------- launch ----------------
extern "C" void kernel_launch(void* const* d_in, const int* in_sizes, int n_in,
                              void* d_out, int out_size, void* d_ws, size_t ws_size,
                              hipStream_t stream) {
    const float* x         = (const float*)d_in[0];
    const float* theta     = (const float*)d_in[1];
    const float* B_re      = (const float*)d_in[2];
    const float* B_im      = (const float*)d_in[3];
    const float* C_re      = (const float*)d_in[4];
    const float* C_im      = (const float*)d_in[5];
    const float* alpha_raw = (const float*)d_in[6];
    const float* beta      = (const float*)d_in[7];
    const float* log_dc    = (const float*)d_in[8];
    const float* log_tau   = (const float*)d_in[9];
    const float* log_scale = (const float*)d_in[10];
    const float* log_bw    = (const float*)d_in[11];
    float* y = (float*)d_out;

    float*  stats = (float*)d_ws;
    float2* Xf    = (float2*)((char*)d_ws + 256);
    float2* Yf    = Xf + (size_t)BATCH * C_IN * NF;

    stats_kernel<<<BATCH, 256, 0, stream>>>(x, stats);
    rowfft_fwd_kernel<<<BATCH * C_IN * H, 128, 0, stream>>>(x, stats, Xf);
    colfft_kernel<<<BATCH * C_IN * WTILES, 256, 0, stream>>>(Xf, -1.0f, 1.0f);
    spectral_kernel<<<(NF / 16) / 8, 256, 0, stream>>>(
        Xf, Yf, theta, B_re, B_im, C_re, C_im,
        alpha_raw, beta, log_dc, log_tau, log_scale, log_bw);
    colfft_kernel<<<BATCH * K_OUT * WTILES, 256, 0, stream>>>(Yf, +1.0f, 1.0f / 384.0f);
    rowfft_inv_kernel<<<BATCH * K_OUT * H, 128, 0, stream>>>(Yf, y);
}